// GCN_64768106823756
// MI455X (gfx1250) — compile-verified
//
#include <hip/hip_runtime.h>
#include <cstdint>
#include <cstddef>

#define FDIM 128
#define NGRAPH 256

typedef float v2f __attribute__((ext_vector_type(2)));
typedef float v8f __attribute__((ext_vector_type(8)));

// ---------------- degree / normalization ----------------
__global__ void k_init_deg(int* deg, int n) {
  int i = blockIdx.x * blockDim.x + threadIdx.x;
  if (i < n) deg[i] = 1;  // self loop contributes 1
}

__global__ void k_edge_deg(const int* __restrict__ dst, int nE, int* deg) {
  int e = blockIdx.x * blockDim.x + threadIdx.x;
  if (e < nE) atomicAdd(&deg[dst[e]], 1);
}

__global__ void k_dinv(const int* __restrict__ deg, float* __restrict__ dinv, int n) {
  int i = blockIdx.x * blockDim.x + threadIdx.x;
  if (i < n) dinv[i] = rsqrtf((float)deg[i]);  // deg >= 1 always (self loops)
}

// ---------------- propagation ----------------
// self-loop term also initializes the output buffer: hout = hin * dinv^2
// vectorized: one float4 per thread (32 float4 per node row)
__global__ void k_hop_self(const float4* __restrict__ hin, float4* __restrict__ hout,
                           const float* __restrict__ dinv, long total4) {
  long i = (long)blockIdx.x * blockDim.x + threadIdx.x;
  if (i < total4) {
    int node = (int)(i >> 5);  // 128/4 = 32 float4 per row
    float dv = dinv[node];
    float sc = dv * dv;
    float4 v = hin[i];
    v.x *= sc; v.y *= sc; v.z *= sc; v.w *= sc;
    hout[i] = v;
  }
}

// one wave (32 lanes) per edge; each lane owns a contiguous float4 of the row
__global__ void k_hop_edges(const float* __restrict__ hin, float* __restrict__ hout,
                            const int* __restrict__ src, const int* __restrict__ dst,
                            const float* __restrict__ dinv, int nE) {
  int wid  = (int)((blockIdx.x * blockDim.x + threadIdx.x) >> 5);
  int lane = threadIdx.x & 31;
  if (wid >= nE) return;  // wave-uniform
  int s = src[wid], d = dst[wid];
  float nrm = dinv[s] * dinv[d];
  const float4* in = (const float4*)(hin + (size_t)s * FDIM);
  float*       out = hout + (size_t)d * FDIM + lane * 4;
  float4 v = in[lane];
  unsafeAtomicAdd(&out[0], v.x * nrm);
  unsafeAtomicAdd(&out[1], v.y * nrm);
  unsafeAtomicAdd(&out[2], v.z * nrm);
  unsafeAtomicAdd(&out[3], v.w * nrm);
}

// ---------------- dense layer: O = relu(H @ W + b), fp32 WMMA ----------------
// W (128x128, 64KB) staged in LDS via async global->LDS copies (ASYNCcnt),
// one wave per 16x16 output tile; K=128 in 32 steps of V_WMMA_F32_16X16X4_F32
__global__ void k_gemm_bias_relu(const float* __restrict__ H, const float* __restrict__ W,
                                 const float* __restrict__ bias, float* __restrict__ O, int M) {
  __shared__ __align__(16) float Wlds[FDIM * FDIM];  // 64 KB of the WGP's 320 KB
  int tid = threadIdx.x;

  // async-stage W: 16384 floats = 4096 x b128; 256 threads x 16 chunks, coalesced
  {
    const char* gsrc = (const char*)W;
#pragma unroll
    for (int it = 0; it < 16; ++it) {
      int byteoff = (tid + it * 256) * 16;
      unsigned ldsa = (unsigned)(size_t)((char*)Wlds + byteoff);
      const char* ga = gsrc + byteoff;
      asm volatile("global_load_async_to_lds_b128 %0, %1, off"
                   :: "v"(ldsa), "v"(ga) : "memory");
    }
    asm volatile("s_wait_asynccnt 0x0" ::: "memory");
    __syncthreads();
  }

  int wid  = blockIdx.x * 8 + (tid >> 5);
  int lane = tid & 31;
  int tm = wid >> 3;        // 128/16 = 8 column tiles
  int tn = wid & 7;
  if (tm * 16 >= M) return; // wave-uniform: EXEC all-ones through the WMMAs
  int m0 = tm * 16, n0 = tn * 16;
  int half = lane >> 4;     // 0 -> lanes 0-15, 1 -> lanes 16-31
  int l16  = lane & 15;
  int arow = m0 + l16; if (arow >= M) arow = M - 1;
  const float* Ap = H + (size_t)arow * FDIM + 2 * half;      // A: row=l16, K=k+2*half+{0,1}
  const float* Bl = Wlds + n0 + l16 + 2 * half * FDIM;       // B: reg r -> K=k+r+2*half, N=l16
  v8f acc = {0.f, 0.f, 0.f, 0.f, 0.f, 0.f, 0.f, 0.f};
#pragma unroll
  for (int k = 0; k < FDIM; k += 4) {
    v2f a = *(const v2f*)(Ap + k);
    v2f b;
    b.x = Bl[k * FDIM];
    b.y = Bl[k * FDIM + FDIM];
    acc = __builtin_amdgcn_wmma_f32_16x16x4_f32(false, a, false, b, (short)0, acc, false, false);
  }
  float bv = bias[n0 + l16];
#pragma unroll
  for (int r = 0; r < 8; ++r) {           // D: reg r -> M = m0 + r + 8*half, N = n0 + l16
    int m = m0 + r + 8 * half;
    if (m < M) {
      float v = acc[r] + bv;
      O[(size_t)m * FDIM + n0 + l16] = v > 0.f ? v : 0.f;
    }
  }
}

// ---------------- mean pool + log_softmax ----------------
__global__ void k_zero_pool(float* sums, int* counts) {
  int i = blockIdx.x * blockDim.x + threadIdx.x;
  if (i < NGRAPH * FDIM) sums[i] = 0.f;
  if (i < NGRAPH) counts[i] = 0;
}

__global__ void k_pool(const float* __restrict__ h, const int* __restrict__ batch,
                       float* sums, int* counts, int n) {
  int wid  = (int)((blockIdx.x * blockDim.x + threadIdx.x) >> 5);
  int lane = threadIdx.x & 31;
  if (wid >= n) return;
  int g = batch[wid];
  const float4* in = (const float4*)(h + (size_t)wid * FDIM);
  float*       out = sums + (size_t)g * FDIM + lane * 4;
  float4 v = in[lane];
  unsafeAtomicAdd(&out[0], v.x);
  unsafeAtomicAdd(&out[1], v.y);
  unsafeAtomicAdd(&out[2], v.z);
  unsafeAtomicAdd(&out[3], v.w);
  if (lane == 0) atomicAdd(&counts[g], 1);
}

__global__ void k_logsoftmax(const float* __restrict__ sums, const int* __restrict__ counts,
                             float* __restrict__ out) {
  __shared__ float red[FDIM];
  int g = blockIdx.x, t = threadIdx.x;  // 128 threads per graph
  float cnt = (float)counts[g]; if (cnt < 1.f) cnt = 1.f;
  float v = sums[g * FDIM + t] / cnt;
  red[t] = v; __syncthreads();
  for (int s = 64; s > 0; s >>= 1) { if (t < s) red[t] = fmaxf(red[t], red[t + s]); __syncthreads(); }
  float m = red[0]; __syncthreads();
  red[t] = expf(v - m); __syncthreads();
  for (int s = 64; s > 0; s >>= 1) { if (t < s) red[t] += red[t + s]; __syncthreads(); }
  float lse = m + logf(red[0]);
  out[g * FDIM + t] = v - lse;
}

// ---------------- host ----------------
extern "C" void kernel_launch(void* const* d_in, const int* in_sizes, int n_in,
                              void* d_out, int out_size, void* d_ws, size_t ws_size,
                              hipStream_t stream) {
  (void)n_in; (void)out_size; (void)ws_size;
  const float* x     = (const float*)d_in[0];
  const int*   ei    = (const int*)d_in[1];
  const int*   batch = (const int*)d_in[2];
  const float* W     = (const float*)d_in[3];
  const float* bias  = (const float*)d_in[4];
  float*       outp  = (float*)d_out;

  int n  = in_sizes[0] / FDIM;   // 100000
  int nE = in_sizes[1] / 2;      // 1600000
  const int* src = ei;           // edge_index[0]
  const int* dst = ei + nE;      // edge_index[1]

  auto al = [](size_t o) { return (o + 511) & ~(size_t)511; };
  char* ws = (char*)d_ws;
  size_t o = 0;
  int*   deg    = (int*)(ws + o);   o = al(o + (size_t)n * 4);
  float* dinv   = (float*)(ws + o); o = al(o + (size_t)n * 4);
  float* hA     = (float*)(ws + o); o = al(o + (size_t)n * FDIM * 4);
  float* hB     = (float*)(ws + o); o = al(o + (size_t)n * FDIM * 4);
  float* sums   = (float*)(ws + o); o = al(o + (size_t)NGRAPH * FDIM * 4);
  int*   counts = (int*)(ws + o);   o = al(o + (size_t)NGRAPH * 4);

  const int tpb = 256;  // 8 wave32s per block
  long total4 = (long)n * (FDIM / 4);

  k_init_deg<<<(n + tpb - 1) / tpb, tpb, 0, stream>>>(deg, n);
  k_edge_deg<<<(nE + tpb - 1) / tpb, tpb, 0, stream>>>(dst, nE, deg);
  k_dinv<<<(n + tpb - 1) / tpb, tpb, 0, stream>>>(deg, dinv, n);

  // hop 1: x -> hA
  k_hop_self<<<(int)((total4 + tpb - 1) / tpb), tpb, 0, stream>>>(
      (const float4*)x, (float4*)hA, dinv, total4);
  k_hop_edges<<<(nE + 7) / 8, tpb, 0, stream>>>(x, hA, src, dst, dinv, nE);
  // hop 2: hA -> hB
  k_hop_self<<<(int)((total4 + tpb - 1) / tpb), tpb, 0, stream>>>(
      (const float4*)hA, (float4*)hB, dinv, total4);
  k_hop_edges<<<(nE + 7) / 8, tpb, 0, stream>>>(hA, hB, src, dst, dinv, nE);

  // dense: relu(hB @ W + b) -> hA
  int rowTiles = (n + 15) / 16;
  int waves = rowTiles * 8;
  k_gemm_bias_relu<<<(waves + 7) / 8, tpb, 0, stream>>>(hB, W, bias, hA, n);

  // pool + log_softmax
  k_zero_pool<<<(NGRAPH * FDIM + tpb - 1) / tpb, tpb, 0, stream>>>(sums, counts);
  k_pool<<<(n + 7) / 8, tpb, 0, stream>>>(hA, batch, sums, counts, n);
  k_logsoftmax<<<NGRAPH, FDIM, 0, stream>>>(sums, counts, outp);
}